// SO3Linear_49546742727010
// MI455X (gfx1250) — compile-verified
//
#include <hip/hip_runtime.h>

typedef __attribute__((ext_vector_type(2))) float v2f;
typedef __attribute__((ext_vector_type(4))) float v4f;
typedef __attribute__((ext_vector_type(8))) float v8f;

#define NUM_M   25
#define IN_F    128
#define OUT_F   128
#define TILE_B  16
#define A_STRIDE 132   // padded: row base bank = 4*r, 16B-aligned rows

__global__ __launch_bounds__(256)
void so3linear_wmma_f32(const float* __restrict__ x,
                        const float* __restrict__ weight,
                        const float* __restrict__ bias,
                        float* __restrict__ out,
                        float bound) {
    __shared__ float As[TILE_B * A_STRIDE];

    const int m  = blockIdx.y;          // 0..24 (uniform per block)
    const int b0 = blockIdx.x * TILE_B; // batch tile base

    // degree l such that m lies in the 2l+1 block (uniform scalar loop)
    int l = 0, mm = m;
    while (mm >= 2 * l + 1) { mm -= 2 * l + 1; ++l; }

    const int tid = threadIdx.x;

    // ---- cooperative load of A tile: 16 batch rows x 128 in-features ----
    {
        const int r  = tid >> 4;           // 0..15
        const int kb = (tid & 15) << 3;    // 0,8,...,120
        const float* src = x + ((size_t)(b0 + r) * NUM_M + m) * IN_F + kb;
        v4f p0 = *(const v4f*)(src);
        v4f p1 = *(const v4f*)(src + 4);
        *(v4f*)(&As[r * A_STRIDE + kb])     = p0;
        *(v4f*)(&As[r * A_STRIDE + kb + 4]) = p1;
    }
    __syncthreads();

    const int wave = tid >> 5;   // 0..7 -> output-col tile base n0 = 16*wave
    const int lane = tid & 31;
    const int c    = lane & 15;  // col within tile (also A row index)
    const int g    = lane >> 4;  // lane-group: K pair select / M+8 select

    // B fragment source: weight[l][n0+c][k], row-major over i
    const float* wrow = weight + ((size_t)l * OUT_F + (wave * 16 + c)) * IN_F;

    v8f acc = {0.f, 0.f, 0.f, 0.f, 0.f, 0.f, 0.f, 0.f};

    // ---- K loop: 32 x V_WMMA_F32_16X16X4_F32 ----
    #pragma unroll
    for (int kk = 0; kk < IN_F; kk += 4) {
        // A 16x4: lane holds A[row=c][kk+2g .. kk+2g+1]  (ds_load_b64)
        v2f a = *(const v2f*)(&As[c * A_STRIDE + kk + 2 * g]);
        // B 4x16: lane holds B[kk+2g .. +1][col=c] = w[n0+c][kk+2g .. +1]
        v2f braw = *(const v2f*)(wrow + kk + 2 * g);
        v2f b;
        b.x = braw.x - bound;
        b.y = braw.y - bound;
        acc = __builtin_amdgcn_wmma_f32_16x16x4_f32(
                  /*neg_a=*/false, a, /*neg_b=*/false, b,
                  /*c_mod=*/(short)0, acc, /*reuse_a=*/false, /*reuse_b=*/false);
    }

    // bias only on the l=0 (m==0) component; uniform branch
    if (m == 0) {
        float bv = bias[wave * 16 + c];
        #pragma unroll
        for (int v = 0; v < 8; ++v) acc[v] += bv;
    }

    // D layout: acc[v] = D[v + 8*g][c]
    #pragma unroll
    for (int v = 0; v < 8; ++v) {
        out[((size_t)(b0 + v + 8 * g) * NUM_M + m) * OUT_F + wave * 16 + c] = acc[v];
    }
}

extern "C" void kernel_launch(void* const* d_in, const int* in_sizes, int n_in,
                              void* d_out, int out_size, void* d_ws, size_t ws_size,
                              hipStream_t stream) {
    const float* x      = (const float*)d_in[0];  // [16384, 25, 128]
    const float* weight = (const float*)d_in[1];  // [5, 128, 128]
    const float* bias   = (const float*)d_in[2];  // [128]
    float*       out    = (float*)d_out;          // [16384, 25, 128]

    const int batch = in_sizes[0] / (NUM_M * IN_F);     // 16384
    const float bound = 0.08838834764831845f;           // 1/sqrt(128)

    dim3 grid(batch / TILE_B, NUM_M);
    dim3 block(256);
    so3linear_wmma_f32<<<grid, block, 0, stream>>>(x, weight, bias, out, bound);
}